// CrossMambaBlock_1915555414511
// MI455X (gfx1250) — compile-verified
//
#include <hip/hip_runtime.h>
#include <math.h>

// ---------------- problem constants (match reference) ----------------
#define Bd   8
#define Nt   1024
#define Ma   512
#define Dm   256
#define Hh   4
#define HD   64
#define DST  8
#define DTR  16

typedef __attribute__((ext_vector_type(2))) float v2f;
typedef __attribute__((ext_vector_type(8))) float v8f;

// ---------------- input index map (setup_inputs recursive dict order) ----
// 0:x 1:ln0_w 2:ln0_b
// block0 @3: in_proj=+0, fwd=+1..+7 (conv_w,conv_b,x_proj_w,dt_proj_w,dt_proj_b,A_log,D),
//            bwd=+8..+14, norm_w=+15, norm_b=+16, out_proj=+17
// block1 @21 (same layout)
// 39:cat0_w 40:cat1_w
// cross: q@41 k@47 v0@53 v1@59 mh0@65 mh1@71  (w,b,gamma,beta,mean,var)
// 77:map_w 78:unmap_w 79:proj_w 80:proj_b

// ---------------- generic strided WMMA-f32 GEMM -----------------------
// C[m,n] = epilogue( alpha * sum_k A(m,k)*B(k,n) )
// A(m,k) = A[z_off + m*a_sm + k*a_sk]   (optional fused softmax on load)
// B(k,n) = B[z_off + k*b_sk + n*b_sn]
// batch z: offset = (z/zdiv)*hi + (z%zdiv)*lo   (per operand)
// One wave computes a 64(M) x 16(N) tile: B float2 reused by 4 WMMAs.
struct GP {
  int Mdim, Ndim, Kdim;
  const float* A; long long a_hi, a_lo; int a_sm, a_sk;
  const float* B; long long b_hi, b_lo; int b_sk, b_sn;
  float*       C; long long c_hi, c_lo; int c_ld;
  const float* R; long long r_hi, r_lo;                 // residual (ld = c_ld)
  const float* bias; const float* gamma; const float* beta;
  const float* mean; const float* var;
  const float* smx; const float* sinv; long long s_hi, s_lo; // row softmax stats
  float alpha; int zdiv;
};

// EPI: 0 none, 2 +resid, 3 +bias+resid, 4 CBR(bn+relu), 5 resid+CBR, 6 softplus(+bias)
// VA/VB: K-stride==1 -> vectorized float2 operand loads
template<int EPI, int ASOFT, int VA, int VB>
__global__ __launch_bounds__(32) void gemm_k(GP p) {
  int z  = blockIdx.z;
  int zq = z / p.zdiv, zr = z - zq * p.zdiv;
  const float* Ab = p.A + (long long)zq * p.a_hi + (long long)zr * p.a_lo;
  const float* Bb = p.B + (long long)zq * p.b_hi + (long long)zr * p.b_lo;
  float*       Cb = p.C + (long long)zq * p.c_hi + (long long)zr * p.c_lo;

  int tm = blockIdx.y << 6, tn = blockIdx.x << 4;    // 64 x 16 per wave
  int lane = threadIdx.x & 31, lhalf = lane >> 4, l16 = lane & 15;
  int kb = lhalf << 1;          // lanes 0-15: K=k0,k0+1 ; lanes 16-31: K=k0+2,k0+3
  int bn = tn + l16;            // B col this lane loads

  const float* Ap[4];
  float rm[4], ri[4];
#pragma unroll
  for (int t = 0; t < 4; ++t) {
    int am = tm + (t << 4) + l16;                    // A row this lane loads
    Ap[t] = Ab + (long long)am * p.a_sm;
    if (ASOFT) {
      rm[t] = (p.smx  + (long long)zq * p.s_hi + (long long)zr * p.s_lo)[am];
      ri[t] = (p.sinv + (long long)zq * p.s_hi + (long long)zr * p.s_lo)[am];
    } else { rm[t] = 0.f; ri[t] = 1.f; }
  }
  const float* Bp = Bb + (long long)bn * p.b_sn;

  v8f acc[4];
#pragma unroll
  for (int t = 0; t < 4; ++t) acc[t] = (v8f){0.f,0.f,0.f,0.f,0.f,0.f,0.f,0.f};

  for (int k0 = 0; k0 < p.Kdim; k0 += 4) {
    int k = k0 + kb;
    v2f bv;
    if (VB) bv = *reinterpret_cast<const v2f*>(Bp + k);
    else { bv.x = Bp[(long long)k * p.b_sk]; bv.y = Bp[(long long)(k + 1) * p.b_sk]; }
#pragma unroll
    for (int t = 0; t < 4; ++t) {
      v2f av;
      if (VA) av = *reinterpret_cast<const v2f*>(Ap[t] + k);
      else { av.x = Ap[t][(long long)k * p.a_sk]; av.y = Ap[t][(long long)(k + 1) * p.a_sk]; }
      if (ASOFT) { av.x = expf(av.x - rm[t]) * ri[t]; av.y = expf(av.y - rm[t]) * ri[t]; }
      acc[t] = __builtin_amdgcn_wmma_f32_16x16x4_f32(false, av, false, bv,
                                                     (short)0, acc[t], false, false);
    }
  }

  int col = tn + l16;
  const float* Rb = (EPI==2 || EPI==3 || EPI==5)
      ? (p.R + (long long)zq * p.r_hi + (long long)zr * p.r_lo) : nullptr;
#pragma unroll
  for (int t = 0; t < 4; ++t) {
#pragma unroll
    for (int r = 0; r < 8; ++r) {
      int row = tm + (t << 4) + r + (lhalf << 3);    // C/D: VGPR r, +8 for lanes>=16
      long long ci = (long long)row * p.c_ld + col;
      float v = acc[t][r] * p.alpha;
      if (EPI == 2)      v += Rb[ci];
      else if (EPI == 3) v += p.bias[col] + Rb[ci];
      else if (EPI == 4 || EPI == 5) {
        float h = v + p.bias[col];
        h = (h - p.mean[col]) * rsqrtf(p.var[col] + 1e-5f) * p.gamma[col] + p.beta[col];
        h = fmaxf(h, 0.f);
        v = (EPI == 5) ? (Rb[ci] + h) : h;
      } else if (EPI == 6) {
        float h = v + p.bias[col];
        v = (h > 20.f) ? h : log1pf(expf(h));
      }
      Cb[ci] = v;
    }
  }
}

static void gemm_go(hipStream_t st, int epi, int asoft, const GP& p, int nz) {
  dim3 g(p.Ndim / 16, p.Mdim / 64, nz), b(32, 1, 1);
  bool va = (p.a_sk == 1), vb = (p.b_sk == 1);
  if (asoft) {
    if (va) gemm_k<0,1,1,0><<<g,b,0,st>>>(p);
    else    gemm_k<0,1,0,0><<<g,b,0,st>>>(p);
    return;
  }
  switch (epi) {
    case 0:
      if (va && vb)      gemm_k<0,0,1,1><<<g,b,0,st>>>(p);
      else if (va)       gemm_k<0,0,1,0><<<g,b,0,st>>>(p);
      else               gemm_k<0,0,0,0><<<g,b,0,st>>>(p);
      break;
    case 2: gemm_k<2,0,1,1><<<g,b,0,st>>>(p); break;
    case 3: gemm_k<3,0,1,1><<<g,b,0,st>>>(p); break;
    case 4: gemm_k<4,0,1,1><<<g,b,0,st>>>(p); break;
    case 5: gemm_k<5,0,1,1><<<g,b,0,st>>>(p); break;
    case 6: gemm_k<6,0,1,1><<<g,b,0,st>>>(p); break;
  }
}

static GP g_basic(int M, int N, int K, const float* A, int asm_, int ask,
                  const float* B, int bsk, int bsn, float* C, int cld,
                  float alpha = 1.f) {
  GP p{}; p.Mdim=M; p.Ndim=N; p.Kdim=K;
  p.A=A; p.a_sm=asm_; p.a_sk=ask;
  p.B=B; p.b_sk=bsk; p.b_sn=bsn;
  p.C=C; p.c_ld=cld; p.alpha=alpha; p.zdiv=1;
  return p;
}

// ---------------- aux device helpers ----------------------------------
__device__ __forceinline__ float wv_sum(float v){
  for (int m = 16; m >= 1; m >>= 1) v += __shfl_xor(v, m, 32);
  return v;
}
__device__ __forceinline__ float wv_max(float v){
  for (int m = 16; m >= 1; m >>= 1) v = fmaxf(v, __shfl_xor(v, m, 32));
  return v;
}

// layernorm over 256 channels; optional fused 0.5*(in1 + flip(in2)) average
__global__ __launch_bounds__(256) void k_layernorm(const float* __restrict__ in1,
    const float* __restrict__ in2, int L, const float* __restrict__ w,
    const float* __restrict__ bb, float* __restrict__ out, int rows) {
  int wid = (blockIdx.x * blockDim.x + threadIdx.x) >> 5;
  if (wid >= rows) return;
  int lane = threadIdx.x & 31;
  long long base = (long long)wid * Dm;
  long long base2 = 0;
  if (in2) {
    int l = wid % L, bidx = wid / L;
    base2 = ((long long)bidx * L + (L - 1 - l)) * Dm;
  }
  float v[8], s = 0.f, q = 0.f;
#pragma unroll
  for (int j = 0; j < 8; ++j) {
    int c = j * 32 + lane;
    float t = in1[base + c];
    if (in2) t = 0.5f * (t + in2[base2 + c]);
    v[j] = t; s += t; q += t * t;
  }
  s = wv_sum(s); q = wv_sum(q);
  float mean = s * (1.f / Dm);
  float var  = q * (1.f / Dm) - mean * mean;
  float rs = rsqrtf(var + 1e-5f);
#pragma unroll
  for (int j = 0; j < 8; ++j) {
    int c = j * 32 + lane;
    out[base + c] = (v[j] - mean) * rs * w[c] + bb[c];
  }
}

// causal depthwise conv (k=2) + SiLU, optional sequence flip
__global__ __launch_bounds__(256) void k_conv_silu(const float* __restrict__ xz,
    const float* __restrict__ cw, const float* __restrict__ cb,
    float* __restrict__ xc, int L, int flip) {
  long long idx = (long long)blockIdx.x * blockDim.x + threadIdx.x;
  int i = (int)(idx % Dm);
  int l = (int)((idx / Dm) % L);
  int b = (int)(idx / ((long long)Dm * L));
  int lo = flip ? (L - 1 - l) : l;
  float xcur = xz[((long long)b * L + lo) * (2 * Dm) + i];
  float xprev = 0.f;
  if (l > 0) {
    int lp = flip ? (lo + 1) : (lo - 1);
    xprev = xz[((long long)b * L + lp) * (2 * Dm) + i];
  }
  float t = xprev * cw[i * 2] + xcur * cw[i * 2 + 1] + cb[i];
  xc[idx] = t / (1.f + expf(-t));
}

// selective scan: 8 lanes per channel hold the 8 hidden states;
// fused u*D add and silu(z) gating
__global__ __launch_bounds__(256) void k_ssm(const float* __restrict__ xc,
    const float* __restrict__ dt, const float* __restrict__ xdbl,
    const float* __restrict__ xz, const float* __restrict__ Alog,
    const float* __restrict__ Dp, float* __restrict__ y, int L, int flip) {
  int tid = threadIdx.x;
  int n = tid & 7;
  int i = blockIdx.x * 32 + (tid >> 3);
  int b = blockIdx.y;
  float a = -expf(Alog[i * DST + n]);
  float Di = Dp[i];
  float h = 0.f;
  for (int l = 0; l < L; ++l) {
    long long tb = (long long)b * L + l;
    float u = xc[tb * Dm + i];
    float d = dt[tb * Dm + i];
    float Bn = xdbl[tb * 32 + DTR + n];
    float Cn = xdbl[tb * 32 + DTR + DST + n];
    h = expf(d * a) * h + d * Bn * u;
    float pnl = h * Cn;
    pnl += __shfl_xor(pnl, 1, 32);
    pnl += __shfl_xor(pnl, 2, 32);
    pnl += __shfl_xor(pnl, 4, 32);
    if (n == 0) {
      int lo = flip ? (L - 1 - l) : l;
      float z = xz[((long long)b * L + lo) * (2 * Dm) + Dm + i];
      float sz = z / (1.f + expf(-z));
      y[tb * Dm + i] = (pnl + u * Di) * sz;
    }
  }
}

// softmax over tokens n (axis=1): one thread per (b,m); two passes, online stats
__global__ __launch_bounds__(256) void k_softmax_tokens(float* __restrict__ p,
    int Ltok, int Mc) {
  int t = blockIdx.x * blockDim.x + threadIdx.x;
  int b = t / Mc, m = t % Mc;
  long long base = (long long)b * Ltok * Mc + m;
  float mx = -1e30f, s = 0.f;
  for (int n = 0; n < Ltok; ++n) {
    float v = p[base + (long long)n * Mc];
    if (v > mx) { s = s * expf(mx - v) + 1.f; mx = v; }
    else s += expf(v - mx);
  }
  float inv = 1.f / s;
  for (int n = 0; n < Ltok; ++n) {
    long long ix = base + (long long)n * Mc;
    p[ix] = expf(p[ix] - mx) * inv;
  }
}

// in-place softmax over contiguous rows of length 512 (one wave per row)
__global__ __launch_bounds__(256) void k_softmax_rows(float* __restrict__ p,
    long long rows) {
  long long wid = ((long long)blockIdx.x * blockDim.x + threadIdx.x) >> 5;
  if (wid >= rows) return;
  int lane = threadIdx.x & 31;
  long long base = wid * Ma;
  float v[16], mx = -1e30f;
#pragma unroll
  for (int j = 0; j < 16; ++j) { v[j] = p[base + j * 32 + lane]; mx = fmaxf(mx, v[j]); }
  mx = wv_max(mx);
  float s = 0.f;
#pragma unroll
  for (int j = 0; j < 16; ++j) { v[j] = expf(v[j] - mx); s += v[j]; }
  s = wv_sum(s);
  float inv = 1.f / s;
#pragma unroll
  for (int j = 0; j < 16; ++j) p[base + j * 32 + lane] = v[j] * inv;
}

// per-row softmax stats (max, 1/sumexp) over contiguous length 512
__global__ __launch_bounds__(256) void k_row_stats(const float* __restrict__ p,
    float* __restrict__ rmx, float* __restrict__ rinv, long long rows) {
  long long wid = ((long long)blockIdx.x * blockDim.x + threadIdx.x) >> 5;
  if (wid >= rows) return;
  int lane = threadIdx.x & 31;
  long long base = wid * Ma;
  float v[16], mx = -1e30f;
#pragma unroll
  for (int j = 0; j < 16; ++j) { v[j] = p[base + j * 32 + lane]; mx = fmaxf(mx, v[j]); }
  mx = wv_max(mx);
  float s = 0.f;
#pragma unroll
  for (int j = 0; j < 16; ++j) s += expf(v[j] - mx);
  s = wv_sum(s);
  if (lane == 0) { rmx[wid] = mx; rinv[wid] = 1.f / s; }
}

// per-column softmax stats over n (stride Mc)
__global__ __launch_bounds__(256) void k_col_stats(const float* __restrict__ p,
    float* __restrict__ cmx, float* __restrict__ cinv, int Ltok, int Mc) {
  int t = blockIdx.x * blockDim.x + threadIdx.x;   // t = bh*Mc + m
  int bh = t / Mc, m = t % Mc;
  long long base = (long long)bh * Ltok * Mc + m;
  float mx = -1e30f, s = 0.f;
  for (int n = 0; n < Ltok; ++n) {
    float v = p[base + (long long)n * Mc];
    if (v > mx) { s = s * expf(mx - v) + 1.f; mx = v; }
    else s += expf(v - mx);
  }
  cmx[t] = mx; cinv[t] = 1.f / s;
}

__global__ __launch_bounds__(256) void k_combine(const float* __restrict__ a,
    const float* __restrict__ b, float* __restrict__ c) {
  long long i = (long long)blockIdx.x * blockDim.x + threadIdx.x;
  c[i] = 0.5f * (a[i] + b[i]);
}

// ---------------- host orchestration ----------------------------------
extern "C" void kernel_launch(void* const* d_in, const int* in_sizes, int n_in,
                              void* d_out, int out_size, void* d_ws, size_t ws_size,
                              hipStream_t stream) {
  (void)in_sizes; (void)n_in; (void)out_size; (void)ws_size;
  const float* X = (const float*)d_in[0];
  auto IN = [&](int i){ return (const float*)d_in[i]; };
  float* W = (float*)d_ws;
  size_t o = 0;
  auto alloc = [&](size_t n){ float* r = W + o; o += n; return r; };

  const size_t BN = (size_t)Bd * Nt, BM = (size_t)Bd * Ma;
  float* x0    = alloc(BN * Dm);
  float* out0  = alloc(BN * Dm);
  float* out0n = alloc(BN * Dm);
  float* out1  = alloc(BM * Dm);
  float* out1n = alloc(BM * Dm);
  float* x1    = alloc(BM * Dm);
  float* yf    = alloc(BN * Dm);
  float* yb    = alloc(BN * Dm);
  float* ycomb = alloc(BN * Dm);
  float* xzb   = alloc(BN * 2 * Dm);
  float* xcb   = alloc(BN * Dm);
  float* dtb   = alloc(BN * Dm);
  float* xdbl  = alloc(BN * 32);
  float* smap  = alloc(BN * Ma);             // reused later as s_unmap
  float* qb    = alloc(BN * Dm);
  float* kb    = alloc(BM * Dm);
  float* v0b   = alloc(BM * Dm);
  float* v1b   = alloc(BN * Dm);
  float* qk    = alloc((size_t)Bd * Hh * Nt * Ma);
  float* rmx   = alloc((size_t)Bd * Hh * Nt);
  float* rinv  = alloc((size_t)Bd * Hh * Nt);
  float* cmx   = alloc((size_t)Bd * Hh * Ma);
  float* cinv  = alloc((size_t)Bd * Hh * Ma);
  float* av0   = alloc(BN * Dm);
  float* av1   = alloc(BM * Dm);
  float* out1s = alloc(BN * Dm);
  float* comb  = alloc(BN * Dm);

  // ---- one bimamba block ----
  auto mamba = [&](const float* xin, int L, int pb, float* yout) {
    int rows = Bd * L;
    // xz = xin @ in_proj^T
    GP p = g_basic(rows, 2 * Dm, Dm, xin, Dm, 1, IN(pb), 1, Dm, xzb, 2 * Dm);
    gemm_go(stream, 0, 0, p, 1);
    for (int dir = 0; dir < 2; ++dir) {
      int db = pb + 1 + dir * 7;
      k_conv_silu<<<(rows * Dm) / 256, 256, 0, stream>>>(
          xzb, IN(db + 0), IN(db + 1), xcb, L, dir);
      // xdbl = xc @ x_proj^T   (N=32)
      p = g_basic(rows, 32, Dm, xcb, Dm, 1, IN(db + 2), 1, Dm, xdbl, 32);
      gemm_go(stream, 0, 0, p, 1);
      // dt = softplus(xdbl[:, :16] @ dt_proj^T + dt_b)
      p = g_basic(rows, Dm, DTR, xdbl, 32, 1, IN(db + 3), 1, DTR, dtb, Dm);
      p.bias = IN(db + 4);
      gemm_go(stream, 6, 0, p, 1);
      k_ssm<<<dim3(Dm / 32, Bd), 256, 0, stream>>>(
          xcb, dtb, xdbl, xzb, IN(db + 5), IN(db + 6), dir ? yb : yf, L, dir);
    }
    // ycomb = LN(0.5*(yf + flip(yb)))
    k_layernorm<<<rows / 8, 256, 0, stream>>>(yf, yb, L, IN(pb + 15), IN(pb + 16),
                                              ycomb, rows);
    // yout = ycomb @ out_proj^T
    p = g_basic(rows, Dm, Dm, ycomb, Dm, 1, IN(pb + 17), 1, Dm, yout, Dm);
    gemm_go(stream, 0, 0, p, 1);
  };

  auto cbr = [&](const float* Ain, int rows, int idx, float* Cout, int epi) {
    GP p = g_basic(rows, Dm, Dm, Ain, Dm, 1, IN(idx), 1, Dm, Cout, Dm);
    p.bias = IN(idx + 1); p.gamma = IN(idx + 2); p.beta = IN(idx + 3);
    p.mean = IN(idx + 4); p.var = IN(idx + 5);
    if (epi == 5) p.R = Cout;
    gemm_go(stream, epi, 0, p, 1);
  };

  // ---- phase A: ln0, block0 ----
  k_layernorm<<<BN / 8, 256, 0, stream>>>(X, nullptr, Nt, IN(1), IN(2), x0, (int)BN);
  mamba(x0, Nt, 3, out0);

  // ---- phase B: anchor map + block1 ----
  { GP p = g_basic((int)BN, Ma, Dm, x0, Dm, 1, IN(77), 1, Dm, smap, Ma);
    gemm_go(stream, 0, 0, p, 1); }
  k_softmax_tokens<<<(Bd * Ma) / 256, 256, 0, stream>>>(smap, Nt, Ma);
  { GP p{}; p.Mdim = Ma; p.Ndim = Dm; p.Kdim = Nt; p.alpha = 1.f; p.zdiv = 1;
    p.A = smap; p.a_sm = 1;  p.a_sk = Ma;  p.a_hi = (long long)Nt * Ma;
    p.B = x0;   p.b_sk = Dm; p.b_sn = 1;   p.b_hi = (long long)Nt * Dm;
    p.C = x1;   p.c_ld = Dm;               p.c_hi = (long long)Ma * Dm;
    gemm_go(stream, 0, 0, p, Bd); }
  mamba(x1, Ma, 21, out1);

  // ---- phase C: cat + cross attention ----
  { GP p = g_basic((int)BN, Dm, Dm, out0, Dm, 1, IN(39), 1, Dm, out0n, Dm);
    p.R = x0; gemm_go(stream, 2, 0, p, 1); }
  { GP p = g_basic((int)BM, Dm, Dm, out1, Dm, 1, IN(40), 1, Dm, out1n, Dm);
    p.R = x1; gemm_go(stream, 2, 0, p, 1); }
  cbr(out0n, (int)BN, 41, qb, 4);
  cbr(out1n, (int)BM, 47, kb, 4);
  cbr(out1n, (int)BM, 53, v0b, 4);
  cbr(out0n, (int)BN, 59, v1b, 4);
  // qk[bh,n,m] = (1/8) sum_d q[b,n,h*64+d] * k[b,m,h*64+d]
  { GP p{}; p.Mdim = Nt; p.Ndim = Ma; p.Kdim = HD; p.alpha = 0.125f; p.zdiv = Hh;
    p.A = qb; p.a_sm = Dm; p.a_sk = 1; p.a_hi = (long long)Nt * Dm; p.a_lo = HD;
    p.B = kb; p.b_sk = 1;  p.b_sn = Dm; p.b_hi = (long long)Ma * Dm; p.b_lo = HD;
    p.C = qk; p.c_ld = Ma; p.c_hi = (long long)Hh * Nt * Ma; p.c_lo = (long long)Nt * Ma;
    gemm_go(stream, 0, 0, p, Bd * Hh); }
  k_row_stats<<<(Bd * Hh * Nt) / 8, 256, 0, stream>>>(qk, rmx, rinv,
                                                      (long long)Bd * Hh * Nt);
  k_col_stats<<<(Bd * Hh * Ma) / 256, 256, 0, stream>>>(qk, cmx, cinv, Nt, Ma);
  // av0[b,n,h*64+d] = sum_m softmax_m(qk)[n,m] * v0[b,m,h*64+d]
  { GP p{}; p.Mdim = Nt; p.Ndim = HD; p.Kdim = Ma; p.alpha = 1.f; p.zdiv = Hh;
    p.A = qk;  p.a_sm = Ma; p.a_sk = 1;  p.a_hi = (long long)Hh * Nt * Ma; p.a_lo = (long long)Nt * Ma;
    p.B = v0b; p.b_sk = Dm; p.b_sn = 1;  p.b_hi = (long long)Ma * Dm;      p.b_lo = HD;
    p.C = av0; p.c_ld = Dm;              p.c_hi = (long long)Nt * Dm;      p.c_lo = HD;
    p.smx = rmx; p.sinv = rinv; p.s_hi = (long long)Hh * Nt; p.s_lo = Nt;
    gemm_go(stream, 0, 1, p, Bd * Hh); }
  // av1[b,m,h*64+d] = sum_n softmax_n(qk)[n,m] * v1[b,n,h*64+d]
  { GP p{}; p.Mdim = Ma; p.Ndim = HD; p.Kdim = Nt; p.alpha = 1.f; p.zdiv = Hh;
    p.A = qk;  p.a_sm = 1;  p.a_sk = Ma; p.a_hi = (long long)Hh * Nt * Ma; p.a_lo = (long long)Nt * Ma;
    p.B = v1b; p.b_sk = Dm; p.b_sn = 1;  p.b_hi = (long long)Nt * Dm;      p.b_lo = HD;
    p.C = av1; p.c_ld = Dm;              p.c_hi = (long long)Ma * Dm;      p.c_lo = HD;
    p.smx = cmx; p.sinv = cinv; p.s_hi = (long long)Hh * Ma; p.s_lo = Ma;
    gemm_go(stream, 0, 1, p, Bd * Hh); }
  cbr(av0, (int)BN, 65, out0n, 5);   // out0 += cbr(mh0)
  cbr(av1, (int)BM, 71, out1n, 5);   // out1 += cbr(mh1)

  // ---- phase D: unmap scatter + final proj ----
  { GP p = g_basic((int)BN, Ma, Dm, x0, Dm, 1, IN(78), 1, Dm, smap, Ma);
    gemm_go(stream, 0, 0, p, 1); }
  k_softmax_rows<<<BN / 8, 256, 0, stream>>>(smap, (long long)BN);
  { GP p{}; p.Mdim = Nt; p.Ndim = Dm; p.Kdim = Ma; p.alpha = 1.f; p.zdiv = 1;
    p.A = smap;  p.a_sm = Ma; p.a_sk = 1; p.a_hi = (long long)Nt * Ma;
    p.B = out1n; p.b_sk = Dm; p.b_sn = 1; p.b_hi = (long long)Ma * Dm;
    p.C = out1s; p.c_ld = Dm;             p.c_hi = (long long)Nt * Dm;
    gemm_go(stream, 0, 0, p, Bd); }
  k_combine<<<(BN * Dm) / 256, 256, 0, stream>>>(out0n, out1s, comb);
  { GP p = g_basic((int)BN, Dm, Dm, comb, Dm, 1, IN(79), 1, Dm, (float*)d_out, Dm);
    p.bias = IN(80); p.R = X;
    gemm_go(stream, 3, 0, p, 1); }
}